// CategoricalSampler_38946763440743
// MI455X (gfx1250) — compile-verified
//
#include <hip/hip_runtime.h>
#include <hip/hip_bf16.h>
#include <stdint.h>

#define AS1 __attribute__((address_space(1)))
#define AS3 __attribute__((address_space(3)))

typedef int v4i __attribute__((vector_size(16)));   // matches builtin's expected int4

#if defined(__AMDGCN__) && __has_builtin(__builtin_amdgcn_global_load_async_to_lds_b128)
#define USE_ASYNC_LDS 1
#else
#define USE_ASYNC_LDS 0
#endif

#if __has_builtin(__builtin_amdgcn_s_wait_asynccnt)
#define WAIT_ASYNC(n) __builtin_amdgcn_s_wait_asynccnt(n)
#else
#define WAIT_ASYNC(n) asm volatile("s_wait_asynccnt %0" ::"i"(n) : "memory")
#endif

static constexpr int B_ = 32, S_ = 8, V_ = 128000;
static constexpr int ROWS = B_ * S_;              // 256
static constexpr int SPLIT = 4;                   // blocks per row
static constexpr int CHUNK = V_ / SPLIT;          // 32000 elements per block
static constexpr int CHUNK4 = CHUNK / 4;          // 8000 float4 per block
static constexpr int T_ = 512;                    // threads per block (16 waves)
static constexpr int NSTAGE = (CHUNK4 + T_ - 1) / T_;  // 16 pipelined stages
static constexpr int PIPE = 4;                    // async ring depth (32 KB LDS)

__device__ __forceinline__ uint32_t mix32(uint32_t x) {
  x ^= x >> 16; x *= 0x21f0aaadu;
  x ^= x >> 15; x *= 0x735a2d97u;
  x ^= x >> 15;
  return x;
}

// Gumbel(0,1) noise from a 32-bit hash: u in (0,1), g = -log(-log(u))
__device__ __forceinline__ float gumbel_of(uint32_t h) {
  float u = ((float)(h >> 8) + 0.5f) * 0x1.0p-24f;
  return -__logf(-__logf(u));
}

__global__ __launch_bounds__(T_) void gumbel_argmax_partial_kernel(
    const float* __restrict__ logits, const float* __restrict__ temperature,
    float2* __restrict__ partials) {
  const int tid   = threadIdx.x;
  const int row   = blockIdx.x >> 2;   // SPLIT == 4
  const int split = blockIdx.x & 3;

  const float invT = 1.0f / temperature[0];
  const float4* __restrict__ src =
      reinterpret_cast<const float4*>(logits + (size_t)row * V_) + (size_t)split * CHUNK4;
  const uint32_t ebase = (uint32_t)row * (uint32_t)V_ + (uint32_t)(split * CHUNK);
  const int vbase = split * CHUNK;

  float bestV = __int_as_float(0xff800000);  // -inf
  int   bestI = 0x7fffffff;

  // Within one thread's scan the index is strictly increasing, so strict '>'
  // alone implements the (val desc, idx asc) policy — no tie-break needed here.
  auto consider = [&](float4 x, int i) {
    const uint32_t e = ebase + ((uint32_t)i << 2);
    const int v = vbase + (i << 2);
    float f[4] = {x.x, x.y, x.z, x.w};
#pragma unroll
    for (int c = 0; c < 4; ++c) {
      const float key = fmaf(f[c], invT, gumbel_of(mix32((e + (uint32_t)c) ^ 0x9E3779B9u)));
      if (key > bestV) { bestV = key; bestI = v + c; }
    }
  };

#if USE_ASYNC_LDS
  // 4-deep async global->LDS ring; each lane owns one float4 slot per buffer,
  // so no cross-lane LDS traffic and no __syncthreads in the pipeline.
  __shared__ float4 stage[PIPE][T_];
  auto issue = [&](int s) {
    int i = s * T_ + tid;
    i = (i < CHUNK4) ? i : (CHUNK4 - 1);  // clamp: uniform issue, dup elem is a harmless tie
    __builtin_amdgcn_global_load_async_to_lds_b128(
        (AS1 v4i*)(src + i), (AS3 v4i*)&stage[s & (PIPE - 1)][tid], 0, 0);
  };
#pragma unroll
  for (int k = 0; k < PIPE; ++k) issue(k);
#pragma unroll
  for (int s = 0; s < NSTAGE; ++s) {
    // Async loads complete in order per wave: outstanding after stage s lands
    // is min(NSTAGE, s+PIPE) - (s+1).
    if (s + 3 < NSTAGE)      { WAIT_ASYNC(3); }
    else if (s + 2 < NSTAGE) { WAIT_ASYNC(2); }
    else if (s + 1 < NSTAGE) { WAIT_ASYNC(1); }
    else                     { WAIT_ASYNC(0); }
    int i = s * T_ + tid;
    i = (i < CHUNK4) ? i : (CHUNK4 - 1);
    consider(stage[s & (PIPE - 1)][tid], i);
    if (s + PIPE < NSTAGE) issue(s + PIPE);  // refill the slot just consumed
  }
#else
  // Fallback: straight b128 streaming loads.
  for (int s = 0; s < NSTAGE; ++s) {
    int i = s * T_ + tid;
    i = (i < CHUNK4) ? i : (CHUNK4 - 1);
    consider(src[i], i);
  }
#endif

  // Intra-wave reduction (wave32) with deterministic (val desc, idx asc) tie-break.
#pragma unroll
  for (int m = 16; m > 0; m >>= 1) {
    const float ov = __shfl_xor(bestV, m, 32);
    const int   oi = __shfl_xor(bestI, m, 32);
    if (ov > bestV || (ov == bestV && oi < bestI)) { bestV = ov; bestI = oi; }
  }

  __shared__ float sV[T_ / 32];
  __shared__ int   sI[T_ / 32];
  const int wave = tid >> 5, lane = tid & 31;
  if (lane == 0) { sV[wave] = bestV; sI[wave] = bestI; }
  __syncthreads();
  if (wave == 0) {
    bestV = (lane < (T_ / 32)) ? sV[lane] : __int_as_float(0xff800000);
    bestI = (lane < (T_ / 32)) ? sI[lane] : 0x7fffffff;
#pragma unroll
    for (int m = 16; m > 0; m >>= 1) {
      const float ov = __shfl_xor(bestV, m, 32);
      const int   oi = __shfl_xor(bestI, m, 32);
      if (ov > bestV || (ov == bestV && oi < bestI)) { bestV = ov; bestI = oi; }
    }
    if (lane == 0) partials[blockIdx.x] = make_float2(bestV, __int_as_float(bestI));
  }
}

__global__ void gumbel_argmax_final_kernel(const float2* __restrict__ partials,
                                           float* __restrict__ out) {
  const int r = blockIdx.x * blockDim.x + threadIdx.x;
  if (r < ROWS) {
    float bv = __int_as_float(0xff800000);
    int   bi = 0x7fffffff;
#pragma unroll
    for (int k = 0; k < SPLIT; ++k) {
      const float2 p = partials[r * SPLIT + k];
      const int i = __float_as_int(p.y);
      if (p.x > bv || (p.x == bv && i < bi)) { bv = p.x; bi = i; }
    }
    out[r] = (float)bi;  // sampled vocab index for row r
  }
}

extern "C" void kernel_launch(void* const* d_in, const int* in_sizes, int n_in,
                              void* d_out, int out_size, void* d_ws, size_t ws_size,
                              hipStream_t stream) {
  const float* logits      = (const float*)d_in[0];
  const float* temperature = (const float*)d_in[1];
  float*  out      = (float*)d_out;
  float2* partials = (float2*)d_ws;   // ROWS * SPLIT * 8 bytes = 8 KB

  gumbel_argmax_partial_kernel<<<ROWS * SPLIT, T_, 0, stream>>>(logits, temperature, partials);
  gumbel_argmax_final_kernel<<<(ROWS + 255) / 256, 256, 0, stream>>>(partials, out);
}